// OrderLoss_26036091748815
// MI455X (gfx1250) — compile-verified
//
#include <hip/hip_runtime.h>
#include <hip/hip_bf16.h>

typedef __attribute__((ext_vector_type(2))) float v2f;
typedef __attribute__((ext_vector_type(8))) float v8f;

#define NROW 384
#define DDIM 512
#define M2   768   // 2*NROW

// ---------------------------------------------------------------------------
// Kernel 1: G = X * X^T with X = [rep0; rep1] (logical concat), f32 WMMA.
// One wave32 per 16x16 tile of G; K loop in steps of 4 using
// V_WMMA_F32_16X16X4_F32 (exact f32 tensor path — precision matters because
// exp(G/0.1) amplifies GEMM error 10x in logit space).
// A layout (16x4 f32): lane L -> row M = L&15; VGPR0/1 = K = ((L>>4)*2)+{0,1}
// B layout (4x16 f32): lane L -> col N = L&15; same K striping.
// C/D layout: VGPR v -> row M = v + (L>=16 ? 8:0), col N = L&15.
// ---------------------------------------------------------------------------
__global__ __launch_bounds__(32) void gemm_xxT_wmma(const float* __restrict__ rep0,
                                                    const float* __restrict__ rep1,
                                                    float* __restrict__ G) {
  const int tm   = blockIdx.x * 16;
  const int tn   = blockIdx.y * 16;
  const int lane = threadIdx.x;           // 0..31, EXEC all ones
  const int r    = lane & 15;
  const int kb   = (lane >> 4) << 1;      // 0 or 2

  const int am = tm + r;
  const int bn = tn + r;
  const float* Arow = (am < NROW) ? (rep0 + am * DDIM) : (rep1 + (am - NROW) * DDIM);
  const float* Brow = (bn < NROW) ? (rep0 + bn * DDIM) : (rep1 + (bn - NROW) * DDIM);

  v8f c = {0.f, 0.f, 0.f, 0.f, 0.f, 0.f, 0.f, 0.f};
  for (int k = 0; k < DDIM; k += 4) {
    v2f a = *(const v2f*)(Arow + k + kb);   // 8-byte aligned b64 load
    v2f b = *(const v2f*)(Brow + k + kb);
    c = __builtin_amdgcn_wmma_f32_16x16x4_f32(
        /*neg_a=*/false, a, /*neg_b=*/false, b,
        /*c_mod=*/(short)0, c, /*reuse_a=*/false, /*reuse_b=*/false);
  }

  const int col     = tn + r;
  const int rowBase = tm + ((lane >> 4) << 3);
#pragma unroll
  for (int v = 0; v < 8; ++v) {
    G[(rowBase + v) * M2 + col] = c[v];
  }
}

// ---------------------------------------------------------------------------
// Kernel 2: CLIP per-row partials. partial[i] = -log(pos_i / S_i),
// S_i = sum_{j!=i} exp(G[i,j]/0.1), pos_i = exp(G[i, pair(i)]/0.1).
// loss_clip = 2/768 * sum_i partial[i] (the two _clip_loss calls are row
// permutations of the same G).
// ---------------------------------------------------------------------------
__global__ __launch_bounds__(256) void clip_rows(const float* __restrict__ G,
                                                 float* __restrict__ partial) {
  const int i = blockIdx.x;     // 0..767
  const int t = threadIdx.x;
  __shared__ float red[256];
  const float* row = G + i * M2;
  float s = 0.0f;
  for (int j = t; j < M2; j += 256) {
    if (j != i) s += expf(row[j] * 10.0f);
  }
  red[t] = s;
  __syncthreads();
  for (int off = 128; off > 0; off >>= 1) {
    if (t < off) red[t] += red[t + off];
    __syncthreads();
  }
  if (t == 0) {
    const int pair = (i < NROW) ? (i + NROW) : (i - NROW);
    const float pos = expf(row[pair] * 10.0f);   // literal form: matches ref inf/nan
    partial[i] = -logf(pos / red[0]);
  }
}

// ---------------------------------------------------------------------------
// Kernel 3: RnC per-row partials for feature set f (0=rep0, 1=rep1).
// Uses the diagonal block of G: sqd = sq_i + sq_j - 2*G[i,j].
// denom[k] = sum_{j!=i} (ld[j] >= ld[k]) * exp(logit[j]); e_row[i] zeroed so
// the hot loop is branch-free. partial[f*384+i] = sum_{k!=i}(logit[k]-log(denom)).
// ---------------------------------------------------------------------------
__global__ __launch_bounds__(256) void rnc_rows(const float* __restrict__ G,
                                                const float* __restrict__ labels,
                                                const float* __restrict__ mean,
                                                const float* __restrict__ stdv,
                                                const int*   __restrict__ epoch,
                                                float* __restrict__ partial) {
  const int i    = blockIdx.x;   // 0..383
  const int f    = blockIdx.y;   // 0 or 1
  const int base = f * NROW;
  const int t    = threadIdx.x;

  __shared__ float e_row[NROW];
  __shared__ float ld_row[NROW];
  __shared__ float lg_row[NROW];
  __shared__ float red[256];

  const float inv_std = 1.0f / (stdv[0] + 1e-8f);
  const float lab_i   = (labels[i] - mean[0]) * inv_std;
  const float t_rnc   = 1.0f - 0.9f * ((float)epoch[0] / 100.0f);
  const float inv_t   = 1.0f / t_rnc;
  const float sq_i    = G[(base + i) * M2 + (base + i)];
  const float* Grow   = G + (base + i) * M2 + base;

  for (int j = t; j < NROW; j += 256) {
    const float g    = Grow[j];
    const float sq_j = G[(base + j) * M2 + (base + j)];
    const float sqd  = fmaxf(sq_i + sq_j - 2.0f * g, 0.0f);
    const float dist = (sqd > 0.0f) ? sqrtf(sqd) : 0.0f;
    const float lg   = -dist * inv_t;
    lg_row[j] = lg;
    e_row[j]  = expf(lg);
    ld_row[j] = fabsf(lab_i - (labels[j] - mean[0]) * inv_std);
  }
  __syncthreads();
  if (t == 0) e_row[i] = 0.0f;   // exclude j==i (offdiag) from all denoms
  __syncthreads();

  float acc = 0.0f;
  for (int k = t; k < NROW; k += 256) {
    if (k == i) continue;
    const float thr = ld_row[k];
    float denom = 0.0f;
    for (int j = 0; j < NROW; ++j) {          // LDS broadcast reads
      denom += (ld_row[j] >= thr) ? e_row[j] : 0.0f;
    }
    acc += lg_row[k] - logf(denom);           // denom >= e_row[k] > 0
  }
  red[t] = acc;
  __syncthreads();
  for (int off = 128; off > 0; off >>= 1) {
    if (t < off) red[t] += red[t + off];
    __syncthreads();
  }
  if (t == 0) partial[f * NROW + i] = red[0];
}

// ---------------------------------------------------------------------------
// Kernel 4: deterministic final reduction -> d_out[3]
// ---------------------------------------------------------------------------
__global__ __launch_bounds__(256) void finalize(const float* __restrict__ p_rnc,
                                                const float* __restrict__ p_clip,
                                                float* __restrict__ out) {
  __shared__ float red[256];
  const int t = threadIdx.x;
  float sums[3] = {0.f, 0.f, 0.f};
  for (int i = t; i < NROW; i += 256) sums[0] += p_rnc[i];
  for (int i = t; i < NROW; i += 256) sums[1] += p_rnc[NROW + i];
  for (int i = t; i < M2;   i += 256) sums[2] += p_clip[i];
#pragma unroll
  for (int s = 0; s < 3; ++s) {
    red[t] = sums[s];
    __syncthreads();
    for (int off = 128; off > 0; off >>= 1) {
      if (t < off) red[t] += red[t + off];
      __syncthreads();
    }
    if (t == 0) {
      if (s < 2) out[s] = -red[0] / (float)(NROW * (NROW - 1));
      else       out[2] = 2.0f * red[0] / (float)M2;
    }
    __syncthreads();
  }
}

extern "C" void kernel_launch(void* const* d_in, const int* in_sizes, int n_in,
                              void* d_out, int out_size, void* d_ws, size_t ws_size,
                              hipStream_t stream) {
  const float* rep0  = (const float*)d_in[0];
  const float* rep1  = (const float*)d_in[1];
  const float* lab   = (const float*)d_in[2];
  const float* mean  = (const float*)d_in[3];
  const float* stdv  = (const float*)d_in[4];
  const int*   epoch = (const int*)d_in[5];

  float* G      = (float*)d_ws;           // 768*768 floats = 2.36 MB
  float* p_rnc  = G + (size_t)M2 * M2;    // 768 floats
  float* p_clip = p_rnc + M2;             // 768 floats

  dim3 gGemm(M2 / 16, M2 / 16);           // 48 x 48 tiles
  gemm_xxT_wmma<<<gGemm, 32, 0, stream>>>(rep0, rep1, G);

  clip_rows<<<M2, 256, 0, stream>>>(G, p_clip);

  dim3 gRnc(NROW, 2);
  rnc_rows<<<gRnc, 256, 0, stream>>>(G, lab, mean, stdv, epoch, p_rnc);

  finalize<<<1, 256, 0, stream>>>(p_rnc, p_clip, (float*)d_out);
}